// Devoxelization_57062935495024
// MI455X (gfx1250) — compile-verified
//
#include <hip/hip_runtime.h>
#include <math.h>

#ifndef __has_builtin
#define __has_builtin(x) 0
#endif

#define RDIM  32
#define CDIM  64
#define NPTS  65536
#define BATCH 8
#define TP    16            // points per block in main kernel
#define STRIDE_F 516        // floats per point in LDS stage: 512 + 4 pad (2064B, 16B aligned)

// ---------- CDNA5 async global->LDS helpers (gfx1250) ----------
typedef int v4i __attribute__((__vector_size__(16)));
typedef __attribute__((address_space(1))) v4i as1_v4i;   // global-space 16B vector
typedef __attribute__((address_space(3))) v4i as3_v4i;   // LDS-space 16B vector
typedef __attribute__((address_space(3))) void as3_void;

__device__ __forceinline__ void async_copy_b128(void* lds, const void* gptr) {
#if __has_builtin(__builtin_amdgcn_global_load_async_to_lds_b128)
  __builtin_amdgcn_global_load_async_to_lds_b128(
      (as1_v4i*)gptr, (as3_v4i*)lds, 0, 0);
#else
  unsigned l32 = (unsigned)(unsigned long long)(as3_void*)lds;
  asm volatile("global_load_async_to_lds_b128 %0, %1, off"
               :: "v"(l32), "v"(gptr)
               : "memory");
#endif
}

__device__ __forceinline__ void wait_asynccnt0() {
#if __has_builtin(__builtin_amdgcn_s_wait_asynccnt)
  __builtin_amdgcn_s_wait_asynccnt(0);
#else
  asm volatile("s_wait_asynccnt 0" ::: "memory");
#endif
}

// ---------- ws header: [0..23] f32 per-(b,coord) mins, [24] u32 bits of max ||p||^2 ----------

__global__ __launch_bounds__(256) void k_init(unsigned* hdr) {
  if (threadIdx.x == 0 && blockIdx.x == 0) hdr[24] = 0u;
}

// One block per (b*3+coord): min over N points.
__global__ __launch_bounds__(256) void k_min(const float* __restrict__ pts,
                                             float* __restrict__ hdr_f) {
  __shared__ float red[256];
  const int g = blockIdx.x;                       // 0..23
  const float* src = pts + (size_t)g * NPTS;
  float m = 3.402823466e+38f;
  for (int i = threadIdx.x; i < NPTS; i += 256) m = fminf(m, src[i]);
  red[threadIdx.x] = m;
  __syncthreads();
  for (int s = 128; s > 0; s >>= 1) {
    if (threadIdx.x < s) red[threadIdx.x] = fminf(red[threadIdx.x], red[threadIdx.x + s]);
    __syncthreads();
  }
  if (threadIdx.x == 0) hdr_f[g] = red[0];
}

// Global max over all (b,n) of ||pts - min||^2 (sqrt is monotone -> take sqrt later).
__global__ __launch_bounds__(256) void k_maxsq(const float* __restrict__ pts,
                                               const float* __restrict__ hdr_f,
                                               unsigned* __restrict__ hdr_u) {
  __shared__ float red[256];
  const int total = BATCH * NPTS;
  float mx = 0.0f;
  for (int i = blockIdx.x * 256 + threadIdx.x; i < total; i += gridDim.x * 256) {
    const int b = i >> 16;
    const int n = i & (NPTS - 1);
    const float* pb = pts + (size_t)b * 3 * NPTS + n;
    const float dx = pb[0]        - hdr_f[b * 3 + 0];
    const float dy = pb[NPTS]     - hdr_f[b * 3 + 1];
    const float dz = pb[2 * NPTS] - hdr_f[b * 3 + 2];
    mx = fmaxf(mx, dx * dx + dy * dy + dz * dz);
  }
  red[threadIdx.x] = mx;
  __syncthreads();
  for (int s = 128; s > 0; s >>= 1) {
    if (threadIdx.x < s) red[threadIdx.x] = fmaxf(red[threadIdx.x], red[threadIdx.x + s]);
    __syncthreads();
  }
  if (threadIdx.x == 0) atomicMax(&hdr_u[24], __float_as_uint(red[0]));  // >=0 floats: bit order == uint order
}

// feat[b,c,x,y,z] -> featT[b,x,y,z,c] so a corner gather is one contiguous 256B run.
__global__ __launch_bounds__(256) void k_transpose(const float* __restrict__ feat,
                                                   float* __restrict__ featT) {
  __shared__ float tile[64][33];                  // +1 pad: conflict-free both phases
  const int tid = threadIdx.x;
  const int bx  = blockIdx.x;                     // b*1024 + x*32 + y
  const int y   = bx & 31;
  const int x   = (bx >> 5) & 31;
  const int b   = bx >> 10;
  const int z   = tid & 31;
  const int c0  = tid >> 5;                       // 0..7
#pragma unroll
  for (int cc = 0; cc < 8; ++cc) {
    const int c = c0 * 8 + cc;
    tile[c][z] = feat[((((size_t)b * 64 + c) * 32 + x) * 32 + y) * 32 + z];
  }
  __syncthreads();
  const int c  = tid & 63;
  const int z0 = tid >> 6;                        // 0..3
  float* dst = featT + ((((size_t)b * 32 + x) * 32 + y) * 32) * 64;
#pragma unroll
  for (int zz = 0; zz < 8; ++zz) {
    const int zq = zz * 4 + z0;
    dst[(size_t)zq * 64 + c] = tile[c][zq];
  }
}

// Main: 16 points/block. A: weights+offsets; B: async gather into LDS; C: blend + coalesced store.
__global__ __launch_bounds__(256) void k_main(const float* __restrict__ pts,
                                              const float* __restrict__ featT,
                                              const unsigned* __restrict__ hdr,
                                              float* __restrict__ out) {
  __shared__ float    stage[TP * STRIDE_F];       // 33,024 B
  __shared__ float    wgt[TP][9];                 // padded: conflict-free
  __shared__ unsigned offb[TP][8];                // byte offsets of 8 corners

  const int tid = threadIdx.x;
  const unsigned gp = blockIdx.x * TP;            // global point index base
  const int b  = gp >> 16;
  const int n0 = gp & (NPTS - 1);
  const float* ptsb = pts + (size_t)b * 3 * NPTS;
  const char* gbase = (const char*)(featT + (size_t)b * (RDIM * RDIM * RDIM) * CDIM);

  if (tid < TP * 8) {
    const int p = tid >> 3;
    const int k = tid & 7;                        // bit0->x, bit1->y, bit2->z (0=left,1=right)
    const int n = n0 + p;
    const float* hf = (const float*)hdr;
    const float denom = sqrtf(__uint_as_float(hdr[24])) + 1e-8f;
    const float s = 31.0f / denom;
    const float vx = (ptsb[n]            - hf[b * 3 + 0]) * s;
    const float vy = (ptsb[NPTS + n]     - hf[b * 3 + 1]) * s;
    const float vz = (ptsb[2 * NPTS + n] - hf[b * 3 + 2]) * s;
    int xl = (int)floorf(vx), yl = (int)floorf(vy), zl = (int)floorf(vz);
    int xr = (int)ceilf(vx),  yr = (int)ceilf(vy),  zr = (int)ceilf(vz);
    xl = min(max(xl, 0), 31); yl = min(max(yl, 0), 31); zl = min(max(zl, 0), 31);
    xr = min(max(xr, 0), 31); yr = min(max(yr, 0), 31); zr = min(max(zr, 0), 31);
    const int xs = (k & 1) ? xr : xl;
    const int ys = (k & 2) ? yr : yl;
    const int zs = (k & 4) ? zr : zl;
    const float wx = (k & 1) ? vx : (1.0f - vx);  // reference uses 1-coord, not 1-frac
    const float wy = (k & 2) ? vy : (1.0f - vy);
    const float wz = (k & 4) ? vz : (1.0f - vz);
    wgt[p][k]  = wx * wy * wz;
    offb[p][k] = ((unsigned)(((xs << 5) + ys) << 5) + (unsigned)zs) << 8;  // *C*4 = *256B
  }
  __syncthreads();

  // 2048 async B128 lane-copies: 16 pts x 8 corners x 256B -> LDS stage.
#pragma unroll
  for (int j = 0; j < 8; ++j) {
    const int e = tid + j * 256;
    const int p = e >> 7;
    const int k = (e >> 4) & 7;
    const int q = e & 15;
    const void* g = gbase + offb[p][k] + q * 16;
    void* l = (char*)(stage + p * STRIDE_F) + k * 256 + q * 16;
    async_copy_b128(l, g);
  }
  wait_asynccnt0();
  __syncthreads();

  const int p  = tid & 15;
  const int sc = tid >> 4;
  float wk[8];
#pragma unroll
  for (int k = 0; k < 8; ++k) wk[k] = wgt[p][k];
  const float* st = stage + p * STRIDE_F;
  float* ob = out + (((size_t)b * 64) << 16) + n0 + p;
#pragma unroll
  for (int j = 0; j < 4; ++j) {
    const int c = sc + j * 16;
    float acc = 0.0f;
#pragma unroll
    for (int k = 0; k < 8; ++k) acc = fmaf(wk[k], st[k * 64 + c], acc);
    ob[(size_t)c << 16] = acc;
  }
}

// Fallback if ws can't hold the 64MB transposed grid: direct scattered gathers.
__global__ __launch_bounds__(256) void k_main_fb(const float* __restrict__ pts,
                                                 const float* __restrict__ feat,
                                                 const unsigned* __restrict__ hdr,
                                                 float* __restrict__ out) {
  const float* hf = (const float*)hdr;
  const float denom = sqrtf(__uint_as_float(hdr[24])) + 1e-8f;
  const float s = 31.0f / denom;
  const size_t total = (size_t)BATCH * CDIM * NPTS;
  for (size_t o = (size_t)blockIdx.x * 256 + threadIdx.x; o < total;
       o += (size_t)gridDim.x * 256) {
    const int n = (int)(o & (NPTS - 1));
    const int c = (int)((o >> 16) & 63);
    const int b = (int)(o >> 22);
    const float* pb = pts + (size_t)b * 3 * NPTS + n;
    const float vx = (pb[0]        - hf[b * 3 + 0]) * s;
    const float vy = (pb[NPTS]     - hf[b * 3 + 1]) * s;
    const float vz = (pb[2 * NPTS] - hf[b * 3 + 2]) * s;
    int xl = (int)floorf(vx), yl = (int)floorf(vy), zl = (int)floorf(vz);
    int xr = (int)ceilf(vx),  yr = (int)ceilf(vy),  zr = (int)ceilf(vz);
    xl = min(max(xl, 0), 31); yl = min(max(yl, 0), 31); zl = min(max(zl, 0), 31);
    xr = min(max(xr, 0), 31); yr = min(max(yr, 0), 31); zr = min(max(zr, 0), 31);
    const float* fb = feat + (((size_t)b * 64 + c) << 15);
    const float wxl = 1.0f - vx, wyl = 1.0f - vy, wzl = 1.0f - vz;
    const float f00 = fb[(xl << 10) + (yl << 5) + zl] * wxl + fb[(xr << 10) + (yl << 5) + zl] * vx;
    const float f10 = fb[(xl << 10) + (yr << 5) + zl] * wxl + fb[(xr << 10) + (yr << 5) + zl] * vx;
    const float f01 = fb[(xl << 10) + (yl << 5) + zr] * wxl + fb[(xr << 10) + (yl << 5) + zr] * vx;
    const float f11 = fb[(xl << 10) + (yr << 5) + zr] * wxl + fb[(xr << 10) + (yr << 5) + zr] * vx;
    const float f0 = f00 * wyl + f10 * vy;
    const float f1 = f01 * wyl + f11 * vy;
    out[o] = f0 * wzl + f1 * vz;
  }
}

extern "C" void kernel_launch(void* const* d_in, const int* in_sizes, int n_in,
                              void* d_out, int out_size, void* d_ws, size_t ws_size,
                              hipStream_t stream) {
  const float* pts  = (const float*)d_in[0];   // [8,3,65536] f32
  const float* feat = (const float*)d_in[1];   // [8,64,32,32,32] f32
  float* out = (float*)d_out;                  // [8,64,65536] f32
  unsigned* hdr = (unsigned*)d_ws;

  k_init <<<1, 256, 0, stream>>>(hdr);
  k_min  <<<BATCH * 3, 256, 0, stream>>>(pts, (float*)d_ws);
  k_maxsq<<<256, 256, 0, stream>>>(pts, (const float*)d_ws, hdr);

  const size_t featT_bytes = (size_t)BATCH * CDIM * RDIM * RDIM * RDIM * sizeof(float);
  if (ws_size >= 256 + featT_bytes) {
    float* featT = (float*)((char*)d_ws + 256);
    k_transpose<<<BATCH * RDIM * RDIM, 256, 0, stream>>>(feat, featT);
    k_main<<<(BATCH * NPTS) / TP, 256, 0, stream>>>(pts, featT, hdr, out);
  } else {
    k_main_fb<<<4096, 256, 0, stream>>>(pts, feat, hdr, out);
  }
  (void)in_sizes; (void)n_in; (void)out_size;
}